// MultiHeadGraphAttention_48704929137091
// MI455X (gfx1250) — compile-verified
//
#include <hip/hip_runtime.h>
#include <hip/hip_bf16.h>

// ---------------------------------------------------------------------------
// MultiHeadGraphAttention for MI455X (gfx1250)
//   N=100000 nodes, E=1600000 edges, HIDDEN=64, HEADS=8, HEAD_DIM=8
//   Dense projections via V_WMMA_F32_16X16X4_F32 (exact fp32 matrix path).
//   Edge phase via L2-resident gathers + f32/u32 global atomics.
// ---------------------------------------------------------------------------

typedef __attribute__((ext_vector_type(2))) float v2f;
typedef __attribute__((ext_vector_type(8))) float v8f;

#define HIDDEN 64
#define HEADS 8
#define HEAD_DIM 8

// --------------------------- fp32 64-wide GEMM ------------------------------
// Y[N,64] = X[N,64] @ W[64,64] + b[64]
// Block: 128 threads = 4 waves; each wave computes a 16x64 row tile.
// K=64 decomposed into 16 steps of V_WMMA_F32_16X16X4_F32.
__global__ __launch_bounds__(128) void gemm64_wmma(
    const float* __restrict__ X, const float* __restrict__ W,
    const float* __restrict__ bias, float* __restrict__ Y, int N)
{
    __shared__ float sW[HIDDEN * HIDDEN];
    __shared__ float sB[HIDDEN];

    const int tid = threadIdx.x;

    // Stage W (16 KB) and bias into LDS: 1024 float4 / 128 threads = 8 each.
    {
        const float4* Wv4 = reinterpret_cast<const float4*>(W);
        float4* sWv4 = reinterpret_cast<float4*>(sW);
#pragma unroll
        for (int i = 0; i < 8; ++i)
            sWv4[tid + i * 128] = Wv4[tid + i * 128];
        if (tid < HIDDEN) sB[tid] = bias[tid];
    }
    __syncthreads();

    const int wave   = tid >> 5;        // 0..3
    const int lane   = tid & 31;
    const int lane15 = lane & 15;
    const int hi     = lane >> 4;       // 0 or 1
    const int row_base = blockIdx.x * 64 + wave * 16;

    // A-fragment rows: clamp OOB rows so EXEC stays all-ones for WMMA.
    int r = row_base + lane15;
    if (r >= N) r = N - 1;
    const float* xr = X + (size_t)r * HIDDEN;

    // Preload all 16 A fragments (ISA 16x4 f32 layout):
    //   lane<16: v0=K0,v1=K1 ; lane>=16: v0=K2,v1=K3   (per 4-wide K step)
    v2f a[16];
#pragma unroll
    for (int kt = 0; kt < 16; ++kt) {
        const int k = kt * 4 + hi * 2;
        const float2 t = *reinterpret_cast<const float2*>(xr + k);
        a[kt].x = t.x;
        a[kt].y = t.y;
    }

    // 4 column tiles of 16.
#pragma unroll
    for (int nt = 0; nt < 4; ++nt) {
        const int n = nt * 16 + lane15;
        v8f c = {};
#pragma unroll
        for (int kt = 0; kt < 16; ++kt) {
            const int k = kt * 4 + hi * 2;
            v2f b;
            b.x = sW[(k + 0) * HIDDEN + n];   // B 4x16: mirrors A striping
            b.y = sW[(k + 1) * HIDDEN + n];
            c = __builtin_amdgcn_wmma_f32_16x16x4_f32(
                    /*neg_a=*/false, a[kt], /*neg_b=*/false, b,
                    /*c_mod=*/(short)0, c, /*reuse_a=*/false, /*reuse_b=*/false);
        }
        const float bn = sB[n];
        // C/D layout: vgpr j -> row m = j + 8*hi, col n = lane15.
#pragma unroll
        for (int j = 0; j < 8; ++j) {
            const int m = row_base + j + hi * 8;
            if (m < N) Y[(size_t)m * HIDDEN + n] = c[j] + bn;
        }
    }
}

// ------------------- monotonic float<->uint ordering keys -------------------
__device__ __forceinline__ unsigned fkey(float f) {
    unsigned u = __float_as_uint(f);
    return (u & 0x80000000u) ? ~u : (u | 0x80000000u);
}
__device__ __forceinline__ float funkey(unsigned u) {
    unsigned b = (u & 0x80000000u) ? (u & 0x7FFFFFFFu) : ~u;
    return __uint_as_float(b);
}

// ------------------------- edge scores + segment max ------------------------
// One thread per edge: 8 head dot-products, store raw scores,
// atomicMax (uint-keyed) of the per-edge max into node_max[row].
__global__ __launch_bounds__(256) void edge_scores(
    const int* __restrict__ rows, const int* __restrict__ cols,
    const float* __restrict__ Q, const float* __restrict__ K,
    float* __restrict__ scores, unsigned* __restrict__ node_max, int E)
{
    const int e = blockIdx.x * blockDim.x + threadIdx.x;
    if (e >= E) return;
    const int r = rows[e];
    const int c = cols[e];
    const float4* q4 = reinterpret_cast<const float4*>(Q + (size_t)r * HIDDEN);
    const float4* k4 = reinterpret_cast<const float4*>(K + (size_t)c * HIDDEN);
    const float inv_scale = 0.35355339059327373f;   // 1/sqrt(HEAD_DIM)

    float mx = -3.402823466e38f;
#pragma unroll
    for (int h = 0; h < HEADS; ++h) {
        const float4 qa = q4[2 * h + 0], qb = q4[2 * h + 1];
        const float4 ka = k4[2 * h + 0], kb = k4[2 * h + 1];
        float d = qa.x * ka.x + qa.y * ka.y + qa.z * ka.z + qa.w * ka.w
                + qb.x * kb.x + qb.y * kb.y + qb.z * kb.z + qb.w * kb.w;
        d *= inv_scale;
        scores[(size_t)e * HEADS + h] = d;
        mx = fmaxf(mx, d);
    }
    atomicMax(node_max + r, fkey(mx));
}

// --------------------------- exp + segment sum ------------------------------
// One thread per (edge, head).
__global__ __launch_bounds__(256) void edge_exp(
    const int* __restrict__ rows, const unsigned* __restrict__ node_max,
    float* __restrict__ scores, float* __restrict__ sums, int E)
{
    const long long t = (long long)blockIdx.x * blockDim.x + threadIdx.x;
    const int e = (int)(t >> 3);
    const int h = (int)(t & 7);
    if (e >= E) return;
    const int r = rows[e];
    const float m  = funkey(node_max[r]);
    const float ex = expf(scores[(size_t)e * HEADS + h] - m);
    scores[(size_t)e * HEADS + h] = ex;
    atomicAdd(&sums[(size_t)r * HEADS + h], ex);
}

// ------------------------ attn * V scatter-add ------------------------------
// One thread per (edge, head): 8 f32 atomics into agg[row].
__global__ __launch_bounds__(256) void edge_scatter(
    const int* __restrict__ rows, const int* __restrict__ cols,
    const float* __restrict__ V, const float* __restrict__ scores,
    const float* __restrict__ sums, float* __restrict__ agg, int E)
{
    const long long t = (long long)blockIdx.x * blockDim.x + threadIdx.x;
    const int e = (int)(t >> 3);
    const int h = (int)(t & 7);
    if (e >= E) return;
    const int r = rows[e];
    const int c = cols[e];
    const float attn = scores[(size_t)e * HEADS + h] /
                       (sums[(size_t)r * HEADS + h] + 1e-8f);
    const float4* v4 = reinterpret_cast<const float4*>(
        V + (size_t)c * HIDDEN + h * HEAD_DIM);
    const float4 v0 = v4[0], v1 = v4[1];
    float* out = agg + (size_t)r * HIDDEN + h * HEAD_DIM;
    atomicAdd(out + 0, attn * v0.x);
    atomicAdd(out + 1, attn * v0.y);
    atomicAdd(out + 2, attn * v0.z);
    atomicAdd(out + 3, attn * v0.w);
    atomicAdd(out + 4, attn * v1.x);
    atomicAdd(out + 5, attn * v1.y);
    atomicAdd(out + 6, attn * v1.z);
    atomicAdd(out + 7, attn * v1.w);
}

// ---------------------------------------------------------------------------
extern "C" void kernel_launch(void* const* d_in, const int* in_sizes, int n_in,
                              void* d_out, int out_size, void* d_ws, size_t ws_size,
                              hipStream_t stream)
{
    const float* x  = (const float*)d_in[0];
    const int*   ei = (const int*)  d_in[1];   // [2, E] flattened, int32
    const float* Wq = (const float*)d_in[2];
    const float* bq = (const float*)d_in[3];
    const float* Wk = (const float*)d_in[4];
    const float* bk = (const float*)d_in[5];
    const float* Wv = (const float*)d_in[6];
    const float* bv = (const float*)d_in[7];
    const float* Wo = (const float*)d_in[8];
    const float* bo = (const float*)d_in[9];

    const int N = in_sizes[0] / HIDDEN;
    const int E = in_sizes[1] / 2;
    const int* rows = ei;
    const int* cols = ei + E;

    // Workspace layout (floats unless noted):
    //   Q[N*64] | K[N*64] | V[N*64] | agg[N*64] | node_max(u32)[N] |
    //   sums[N*8] | exp_scores[E*8]
    float* ws   = (float*)d_ws;
    float* Q    = ws;
    float* Kf   = Q  + (size_t)N * HIDDEN;
    float* Vf   = Kf + (size_t)N * HIDDEN;
    float* agg  = Vf + (size_t)N * HIDDEN;
    unsigned* nmax = (unsigned*)(agg + (size_t)N * HIDDEN);
    float* sums = (float*)(nmax + N);
    float* esc  = sums + (size_t)N * HEADS;

    // Zero-init accumulators (memset nodes are graph-capture safe).
    hipMemsetAsync(nmax, 0, (size_t)N * sizeof(unsigned), stream);       // key(min)
    hipMemsetAsync(sums, 0, (size_t)N * HEADS * sizeof(float), stream);
    hipMemsetAsync(agg,  0, (size_t)N * HIDDEN * sizeof(float), stream);

    const dim3 gemm_grid((N + 63) / 64);
    gemm64_wmma<<<gemm_grid, 128, 0, stream>>>(x, Wq, bq, Q,  N);
    gemm64_wmma<<<gemm_grid, 128, 0, stream>>>(x, Wk, bk, Kf, N);
    gemm64_wmma<<<gemm_grid, 128, 0, stream>>>(x, Wv, bv, Vf, N);

    edge_scores<<<(E + 255) / 256, 256, 0, stream>>>(rows, cols, Q, Kf, esc, nmax, E);

    const long long EH = (long long)E * HEADS;
    const int ehBlocks = (int)((EH + 255) / 256);
    edge_exp<<<ehBlocks, 256, 0, stream>>>(rows, nmax, esc, sums, E);
    edge_scatter<<<ehBlocks, 256, 0, stream>>>(rows, cols, Vf, esc, sums, agg, E);

    gemm64_wmma<<<gemm_grid, 128, 0, stream>>>(agg, Wo, bo, (float*)d_out, N);
}